// RefineDetMultiBoxLoss_26585847562271
// MI455X (gfx1250) — compile-verified
//
#include <hip/hip_runtime.h>
#include <math.h>

#define BATCH 64
#define P 16320
#define NCLS 21
#define BLKS_PER_ROW 16
#define ANCH_PER_BLK (P / BLKS_PER_ROW)   // 1020
static_assert(P % BLKS_PER_ROW == 0, "row split must be exact");

typedef unsigned int v4u __attribute__((ext_vector_type(4)));
typedef int          v8i __attribute__((ext_vector_type(8)));
typedef int          v4i __attribute__((ext_vector_type(4)));
// 4-byte-aligned float vectors (conf rows stride 84B -> only dword aligned)
typedef float vf4 __attribute__((ext_vector_type(4), aligned(4)));
typedef float vf2 __attribute__((ext_vector_type(2), aligned(8)));

__device__ __forceinline__ float smooth_l1(float d) {
    float a = fabsf(d);
    return a < 1.0f ? 0.5f * a * a : a - 0.5f;
}

// ---------------------------------------------------------------------------
// Phase 1: one streaming pass over all inputs. Computes per-anchor CE loss,
// pos mask, smooth-L1; writes loss_for_rank; block-reduces partials.
// ---------------------------------------------------------------------------
__global__ void __launch_bounds__(256)
phase1(const float* __restrict__ loc,  const float* __restrict__ conf,
       const float* __restrict__ arm,  const float* __restrict__ loct,
       const int*   __restrict__ conft,
       float* __restrict__ lfr,        // [BATCH*P] loss_for_rank
       float* __restrict__ part_ll,    // [BATCH*BLKS_PER_ROW]
       float* __restrict__ part_pce,   // [BATCH*BLKS_PER_ROW]
       int*   __restrict__ num_pos)    // [BATCH]
{
    const int b   = blockIdx.y;
    const int blk = blockIdx.x;
    const int base = blk * ANCH_PER_BLK;
    const float OBJ_TH = -4.59511985013459f;   // log(0.01/0.99): p_obj > theta

    float ll = 0.0f, pce = 0.0f;
    int np = 0;

    for (int i = threadIdx.x; i < ANCH_PER_BLK; i += 256) {
        const long long idx = (long long)b * P + (base + i);
        // ARM objectness: softmax[1] > theta  <=>  a1 - a0 > log(theta/(1-theta))
        vf2 a = *(const vf2*)(arm + idx * 2);
        const bool obj = (a.y - a.x) > OBJ_TH;
        const int  t   = conft[idx];

        // stable logsumexp over 21 logits (5x b128 + 1 scalar load)
        const float* cp = conf + idx * NCLS;
        vf4 q0 = *(const vf4*)(cp +  0);
        vf4 q1 = *(const vf4*)(cp +  4);
        vf4 q2 = *(const vf4*)(cp +  8);
        vf4 q3 = *(const vf4*)(cp + 12);
        vf4 q4 = *(const vf4*)(cp + 16);
        float c20 = cp[20];
        float m = c20;
        m = fmaxf(m, fmaxf(fmaxf(q0.x, q0.y), fmaxf(q0.z, q0.w)));
        m = fmaxf(m, fmaxf(fmaxf(q1.x, q1.y), fmaxf(q1.z, q1.w)));
        m = fmaxf(m, fmaxf(fmaxf(q2.x, q2.y), fmaxf(q2.z, q2.w)));
        m = fmaxf(m, fmaxf(fmaxf(q3.x, q3.y), fmaxf(q3.z, q3.w)));
        m = fmaxf(m, fmaxf(fmaxf(q4.x, q4.y), fmaxf(q4.z, q4.w)));
        float s = expf(q0.x - m) + expf(q0.y - m) + expf(q0.z - m) + expf(q0.w - m)
                + expf(q1.x - m) + expf(q1.y - m) + expf(q1.z - m) + expf(q1.w - m)
                + expf(q2.x - m) + expf(q2.y - m) + expf(q2.z - m) + expf(q2.w - m)
                + expf(q3.x - m) + expf(q3.y - m) + expf(q3.z - m) + expf(q3.w - m)
                + expf(q4.x - m) + expf(q4.y - m) + expf(q4.z - m) + expf(q4.w - m)
                + expf(c20 - m);
        const float lse = m + logf(s);
        const float ce  = lse - cp[t];          // scalar gather of the true logit

        const bool pos = (t > 0) && obj;
        lfr[idx] = pos ? 0.0f : ce;             // CE >= 0, so float order == uint order
        if (pos) {
            np++; pce += ce;
            vf4 l = *(const vf4*)(loc  + idx * 4);
            vf4 g = *(const vf4*)(loct + idx * 4);
            ll += smooth_l1(l.x - g.x) + smooth_l1(l.y - g.y)
                + smooth_l1(l.z - g.z) + smooth_l1(l.w - g.w);
        }
    }

    __shared__ float sll[256], spc[256];
    __shared__ int   snp[256];
    sll[threadIdx.x] = ll; spc[threadIdx.x] = pce; snp[threadIdx.x] = np;
    __syncthreads();
    for (int off = 128; off > 0; off >>= 1) {
        if (threadIdx.x < off) {
            sll[threadIdx.x] += sll[threadIdx.x + off];
            spc[threadIdx.x] += spc[threadIdx.x + off];
            snp[threadIdx.x] += snp[threadIdx.x + off];
        }
        __syncthreads();
    }
    if (threadIdx.x == 0) {
        part_ll[b * BLKS_PER_ROW + blk]  = sll[0];
        part_pce[b * BLKS_PER_ROW + blk] = spc[0];
        atomicAdd(&num_pos[b], snp[0]);          // int atomics: deterministic
    }
}

// ---------------------------------------------------------------------------
// Phase 2: per batch row. TDM DMA of the 65 KB loss row into LDS, exact
// radix-select of the k-th largest value, tie-exact top-k sum.
// ---------------------------------------------------------------------------
__global__ void __launch_bounds__(1024)
phase2(const float* __restrict__ lfr, const float* __restrict__ part_pce,
       const int* __restrict__ num_pos, float* __restrict__ row_loss_c)
{
    __shared__ float vals[P];                    // 65280 B of the 320 KB WGP LDS
    __shared__ unsigned int hist[256];
    __shared__ unsigned int sh_prefix, sh_r;
    __shared__ float red[1024];
    __shared__ float sh_pce;

    const int b   = blockIdx.x;
    const int tid = threadIdx.x;

    if (tid == 0) {
        float s = 0.0f;
        for (int j = 0; j < BLKS_PER_ROW; ++j) s += part_pce[b * BLKS_PER_ROW + j];
        sh_pce = s;

        // --- Tensor Data Mover: global row -> LDS (async-tensor path) ------
        unsigned long long ga = (unsigned long long)(uintptr_t)(lfr + (long long)b * P);
        unsigned int lds_off  = (unsigned int)(uintptr_t)(&vals[0]); // low 32b = LDS offset
        v4u g0;
        g0.x = 1u;                                            // count=1, user mode
        g0.y = lds_off;                                       // lds_addr
        g0.z = (unsigned int)(ga & 0xFFFFFFFFu);              // global_addr[31:0]
        g0.w = (unsigned int)((ga >> 32) & 0x01FFFFFFu)       // global_addr[56:32]
             | 0x80000000u;                                   // type=2 ("image")
        v8i g1;
        g1[0] = (int)(2u << 16);                  // data_size = 4 bytes; mask=0
        g1[1] = (int)((unsigned)P << 16);         // tensor_dim0[15:0]
        g1[2] = (int)(1u << 16);                  // tensor_dim0 hi=0 | tensor_dim1=1
        g1[3] = (int)((unsigned)P << 16);         // tensor_dim1 hi=0 | tile_dim0=P
        g1[4] = 1;                                // tile_dim1=1 | tile_dim2=0
        g1[5] = P;                                // tensor_dim0_stride[31:0]
        g1[6] = 0;                                // stride0 hi | stride1 lo
        g1[7] = 0;
        v4i g2 = {0, 0, 0, 0};
        v4i g3 = {0, 0, 0, 0};
        v8i g4 = {0, 0, 0, 0, 0, 0, 0, 0};        // 6-arg builtin variant
        __builtin_amdgcn_tensor_load_to_lds(g0, g1, g2, g3, g4, 0);
        __builtin_amdgcn_s_wait_tensorcnt(0);
    }
    __syncthreads();                              // LDS now visible to all waves

    const int npos = num_pos[b];
    const int k = min(3 * npos, P - 1);
    float topk = 0.0f;

    if (k > 0) {
        unsigned int prefix = 0, r = (unsigned int)k;
        for (int shift = 24; shift >= 0; shift -= 8) {
            if (tid < 256) hist[tid] = 0;
            __syncthreads();
            const unsigned int maskk = (shift == 24) ? 0u : (0xFFFFFFFFu << (shift + 8));
            for (int i = tid; i < P; i += 1024) {
                unsigned int u = __float_as_uint(vals[i]);
                if ((u & maskk) == prefix)
                    atomicAdd(&hist[(u >> shift) & 255u], 1u);
            }
            __syncthreads();
            if (tid == 0) {
                unsigned int cum = 0; int bin = 255;
                for (; bin >= 0; --bin) { cum += hist[bin]; if (cum >= r) break; }
                sh_prefix = prefix | ((unsigned int)bin << shift);
                sh_r      = r - (cum - hist[bin]);
            }
            __syncthreads();
            prefix = sh_prefix; r = sh_r;
        }
        // prefix = bits of the k-th largest value; r = copies of it inside top-k
        const unsigned int tbits = prefix;
        float local = 0.0f;
        for (int i = tid; i < P; i += 1024) {
            float v = vals[i];
            if (__float_as_uint(v) > tbits) local += v;
        }
        red[tid] = local;
        __syncthreads();
        for (int off = 512; off > 0; off >>= 1) {
            if (tid < off) red[tid] += red[tid + off];
            __syncthreads();
        }
        topk = red[0] + (float)r * __uint_as_float(tbits);
    }
    if (tid == 0) row_loss_c[b] = sh_pce + topk;
}

// ---------------------------------------------------------------------------
__global__ void __launch_bounds__(256)
finalize(const float* __restrict__ part_ll, const float* __restrict__ row_loss_c,
         const int* __restrict__ num_pos, float* __restrict__ out)
{
    __shared__ float rf[256], rc[256];
    __shared__ int   ri[256];
    const int tid = threadIdx.x;
    float ll = 0.0f;
    for (int i = tid; i < BATCH * BLKS_PER_ROW; i += 256) ll += part_ll[i];
    float lc = 0.0f; int np = 0;
    if (tid < BATCH) { lc = row_loss_c[tid]; np = num_pos[tid]; }
    rf[tid] = ll; rc[tid] = lc; ri[tid] = np;
    __syncthreads();
    for (int off = 128; off > 0; off >>= 1) {
        if (tid < off) {
            rf[tid] += rf[tid + off];
            rc[tid] += rc[tid + off];
            ri[tid] += ri[tid + off];
        }
        __syncthreads();
    }
    if (tid == 0) {
        const float N = (float)ri[0];
        out[0] = rf[0] / N;
        out[1] = rc[0] / N;
    }
}

__global__ void zero_counts(int* __restrict__ num_pos)
{
    if (threadIdx.x < BATCH) num_pos[threadIdx.x] = 0;
}

// ---------------------------------------------------------------------------
extern "C" void kernel_launch(void* const* d_in, const int* in_sizes, int n_in,
                              void* d_out, int out_size, void* d_ws, size_t ws_size,
                              hipStream_t stream)
{
    const float* loc   = (const float*)d_in[0];
    const float* conf  = (const float*)d_in[1];
    const float* arm   = (const float*)d_in[2];
    const float* loct  = (const float*)d_in[3];
    const int*   conft = (const int*)d_in[4];
    float* out = (float*)d_out;

    float* lfr        = (float*)d_ws;                       // BATCH*P floats
    float* part_ll    = lfr + (size_t)BATCH * P;            // 1024 floats
    float* part_pce   = part_ll + BATCH * BLKS_PER_ROW;     // 1024 floats
    float* row_loss_c = part_pce + BATCH * BLKS_PER_ROW;    // 64 floats
    int*   num_pos    = (int*)(row_loss_c + BATCH);         // 64 ints

    zero_counts<<<1, 64, 0, stream>>>(num_pos);
    phase1<<<dim3(BLKS_PER_ROW, BATCH), 256, 0, stream>>>(
        loc, conf, arm, loct, conft, lfr, part_ll, part_pce, num_pos);
    phase2<<<BATCH, 1024, 0, stream>>>(lfr, part_pce, num_pos, row_loss_c);
    finalize<<<1, 256, 0, stream>>>(part_ll, row_loss_c, num_pos, out);
}